// VectorQuantizer_13915694039863
// MI455X (gfx1250) — compile-verified
//
#include <hip/hip_runtime.h>
#include <hip/hip_bf16.h>
#include <math.h>

typedef __attribute__((ext_vector_type(2))) float v2f;
typedef __attribute__((ext_vector_type(8))) float v8f;
typedef __attribute__((ext_vector_type(4))) unsigned int v4u;
typedef __attribute__((ext_vector_type(8))) int v8i;
typedef __attribute__((ext_vector_type(4))) int v4i;

#define VQ_N 131072           // 32*64*64 rows
#define VQ_K 1024             // codebook entries
#define VQ_D 64               // embedding dim
#define VQ_NUMEL 8388608      // N * D
#define LDS_STRIDE 68         // padded row stride (floats); TDM pad 64+4 DWORDs

// ---------------------------------------------------------------------------
// Kernel 1: b_norm[k] = ||emb_k||^2 ; zero counts + sse accumulator
// ---------------------------------------------------------------------------
__global__ void vq_prep(const float* __restrict__ emb,
                        float* __restrict__ bnorm,
                        float* __restrict__ counts,
                        float* __restrict__ sse)
{
    int k = blockIdx.x * blockDim.x + threadIdx.x;       // 1024 threads total
    if (k < VQ_K) {
        const float4* row = reinterpret_cast<const float4*>(emb + k * VQ_D);
        float s = 0.0f;
        #pragma unroll
        for (int i = 0; i < VQ_D / 4; ++i) {
            float4 v = row[i];
            s += v.x * v.x + v.y * v.y + v.z * v.z + v.w * v.w;
        }
        bnorm[k]  = s;
        counts[k] = 0.0f;
    }
    if (blockIdx.x == 0 && threadIdx.x == 0) *sse = 0.0f;
}

// ---------------------------------------------------------------------------
// Kernel 2: distance GEMM via V_WMMA_F32_16X16X4_F32 + per-row argmin.
// Block = 256 threads (8 waves). Wave 0 issues ONE Tensor-Data-Mover DMA that
// stages the whole 256 KB codebook into LDS with 4-DWORD row padding (stride
// 68 floats -> conflict-free B-fragment reads). Each wave then owns a 16-row
// strip of the input and sweeps all 64 code tiles.
// ---------------------------------------------------------------------------
__global__ void __launch_bounds__(256) vq_main(const float* __restrict__ x,
                                               const float* __restrict__ emb,
                                               const float* __restrict__ bnorm,
                                               int*   __restrict__ idx_out,
                                               float* __restrict__ counts)
{
    extern __shared__ float lds[];
    float* se = lds;                         // [VQ_K][LDS_STRIDE] padded codebook
    float* sb = lds + VQ_K * LDS_STRIDE;     // [VQ_K] b_norm

    const int tid = threadIdx.x;

    // ---- TDM: tensor_load_to_lds of the 2-D tile (64 x 1024 f32) ----------
    // D# group0: count=1 | lds_addr | global_addr[56:0] | type=2
    // D# group1: data_size=4B, pad_enable, pad_interval=5 (64 DW),
    //            pad_amount=3 (4 DW)  => LDS row stride = 68 floats.
    if (tid < 32) {                          // one wave issues the DMA once
        unsigned int lds_addr = (unsigned int)(uintptr_t)(void*)se; // LDS aperture low bits
        unsigned long long ga = (unsigned long long)(uintptr_t)emb;
        v4u g0;
        g0[0] = 1u;                                   // count=1, user descriptor
        g0[1] = lds_addr;                             // LDS byte address of tile
        g0[2] = (unsigned int)ga;                     // global_addr[31:0]
        g0[3] = (unsigned int)((ga >> 32) & 0x01FFFFFFu) | (2u << 30); // [56:32]|type=2
        v8i g1;
        g1[0] = (2 << 16)      // data_size = 4 bytes
              | (1 << 20)      // pad_enable
              | (5 << 22)      // pad_interval: 64 DWORDs
              | (3 << 25);     // pad_amount:   4 DWORDs
        g1[1] = (VQ_D << 16);  // tensor_dim0[15:0] = 64 (atomic_barrier_addr = 0)
        g1[2] = (VQ_K << 16);  // tensor_dim0 hi = 0, tensor_dim1[15:0] = 1024
        g1[3] = (VQ_D << 16);  // tensor_dim1 hi = 0, tile_dim0 = 64
        g1[4] = VQ_K;          // tile_dim1 = 1024, tile_dim2 = 0
        g1[5] = VQ_D;          // tensor_dim0_stride[31:0] = 64 elements
        g1[6] = 0;             // stride hi bits, tensor_dim1_stride lo = 0
        g1[7] = 0;
        v4i gz4 = {0, 0, 0, 0};
        v8i gz8 = {0, 0, 0, 0, 0, 0, 0, 0};
        __builtin_amdgcn_tensor_load_to_lds(g0, g1, gz4, gz4, gz8, 0);
        __builtin_amdgcn_s_wait_tensorcnt(0);
    }
    for (int i = tid; i < VQ_K; i += 256) sb[i] = bnorm[i];
    __syncthreads();

    const int wave = tid >> 5;               // 0..7
    const int lane = tid & 31;
    const int half = lane >> 4;              // 0: lanes 0-15, 1: lanes 16-31
    const int l16  = lane & 15;
    const int rowBase = (blockIdx.x * 8 + wave) * 16;

    // A fragments, ISA 16x4 f32 layout:
    //   lanes 0-15 hold M=l16, K = {4c+0, 4c+1}; lanes 16-31 hold K = {4c+2, 4c+3}
    v2f a[16];
    const float* arow = x + (rowBase + l16) * VQ_D + half * 2;
    #pragma unroll
    for (int c = 0; c < 16; ++c) {
        a[c].x = arow[c * 4 + 0];
        a[c].y = arow[c * 4 + 1];
    }

    // Running argmin state in D-matrix layout: VGPR r <-> row (half*8 + r),
    // lane's N column = tile*16 + l16.
    float best[8];
    int   bidx[8];
    #pragma unroll
    for (int r = 0; r < 8; ++r) { best[r] = 3.4e38f; bidx[r] = 0; }

    for (int t = 0; t < VQ_K / 16; ++t) {
        const int n = t * 16 + l16;
        // B fragment, 4x16 f32 layout: lane holds B[K=4c+half*2+{0,1}][N=l16]
        const float* bcol = se + n * LDS_STRIDE + half * 2;
        v8f acc = {};
        #pragma unroll
        for (int c = 0; c < 16; ++c) {
            v2f b;
            b.x = bcol[c * 4 + 0];
            b.y = bcol[c * 4 + 1];
            acc = __builtin_amdgcn_wmma_f32_16x16x4_f32(
                /*neg_a=*/false, a[c], /*neg_b=*/false, b,
                /*c_mod=*/(short)0, acc, /*reuse_a=*/false, /*reuse_b=*/false);
        }
        const float bn = sb[n];
        #pragma unroll
        for (int r = 0; r < 8; ++r) {
            float s = bn - 2.0f * acc[r];    // a_norm/clamp irrelevant for argmin
            if (s < best[r]) { best[r] = s; bidx[r] = n; }
        }
    }

    // Reduce argmin across the 16 lanes (N dimension) within each half-wave.
    // Tie-break on smaller index to match jnp.argmin first-occurrence.
    #pragma unroll
    for (int m = 1; m < 16; m <<= 1) {
        #pragma unroll
        for (int r = 0; r < 8; ++r) {
            float ob = __shfl_xor(best[r], m, 32);
            int   oi = __shfl_xor(bidx[r], m, 32);
            if (ob < best[r] || (ob == best[r] && oi < bidx[r])) {
                best[r] = ob; bidx[r] = oi;
            }
        }
    }

    if (l16 == 0) {
        #pragma unroll
        for (int r = 0; r < 8; ++r) {
            int row = rowBase + half * 8 + r;
            idx_out[row] = bidx[r];
            atomicAdd(&counts[bidx[r]], 1.0f);
        }
    }
}

// ---------------------------------------------------------------------------
// Kernel 3: gather quantized rows, emit quantized_st = x + (q - x), and
// accumulate sum((q - x)^2) with one atomic per block.
// ---------------------------------------------------------------------------
__global__ void __launch_bounds__(256) vq_gather(const float* __restrict__ x,
                                                 const float* __restrict__ emb,
                                                 const int* __restrict__ idx,
                                                 float* __restrict__ out,
                                                 float* __restrict__ sse)
{
    int i = blockIdx.x * blockDim.x + threadIdx.x;   // one float4 per thread
    int e = i * 4;
    int row = e >> 6;
    int col = e & 63;
    int k = idx[row];
    float4 q  = *reinterpret_cast<const float4*>(emb + k * VQ_D + col);
    float4 xi = *reinterpret_cast<const float4*>(x + e);
    float4 d  = make_float4(q.x - xi.x, q.y - xi.y, q.z - xi.z, q.w - xi.w);
    float4 o  = make_float4(xi.x + d.x, xi.y + d.y, xi.z + d.z, xi.w + d.w);
    *reinterpret_cast<float4*>(out + e) = o;

    float s = d.x * d.x + d.y * d.y + d.z * d.z + d.w * d.w;
    #pragma unroll
    for (int m = 16; m >= 1; m >>= 1) s += __shfl_xor(s, m, 32);

    __shared__ float red[8];
    int lane = threadIdx.x & 31, wv = threadIdx.x >> 5;
    if (lane == 0) red[wv] = s;
    __syncthreads();
    if (threadIdx.x == 0) {
        float t = 0.0f;
        #pragma unroll
        for (int w = 0; w < 8; ++w) t += red[w];
        atomicAdd(sse, t);
    }
}

// ---------------------------------------------------------------------------
// Kernel 4: loss = 1.5 * sse / numel ; perplexity = exp(-sum p log(p+1e-10))
// ---------------------------------------------------------------------------
__global__ void vq_final(const float* __restrict__ counts,
                         const float* __restrict__ sse,
                         float* __restrict__ out_scalars)
{
    __shared__ float red[32];
    int t = threadIdx.x;                     // 1024 threads
    float p = counts[t] * (1.0f / (float)VQ_N);
    float h = p * logf(p + 1e-10f);
    #pragma unroll
    for (int m = 16; m >= 1; m >>= 1) h += __shfl_xor(h, m, 32);
    int lane = t & 31, wv = t >> 5;
    if (lane == 0) red[wv] = h;
    __syncthreads();
    if (t == 0) {
        float H = 0.0f;
        #pragma unroll
        for (int w = 0; w < 32; ++w) H += red[w];
        out_scalars[0] = 1.5f * (*sse) * (1.0f / (float)VQ_NUMEL);  // loss
        out_scalars[1] = expf(-H);                                   // perplexity
    }
}

// ---------------------------------------------------------------------------
extern "C" void kernel_launch(void* const* d_in, const int* in_sizes, int n_in,
                              void* d_out, int out_size, void* d_ws, size_t ws_size,
                              hipStream_t stream)
{
    const float* x   = (const float*)d_in[0];   // [32,64,64,64] -> 8388608 f32
    const float* emb = (const float*)d_in[1];   // [1024,64]     -> 65536 f32
    float* out = (float*)d_out;                 // 8388608 quantized_st + loss + perplexity

    float* bnorm  = (float*)d_ws;               // [1024]
    float* counts = bnorm + VQ_K;               // [1024]
    float* sse    = counts + VQ_K;              // [1]
    int*   idx    = (int*)(sse + 1);            // [131072]

    vq_prep<<<4, 256, 0, stream>>>(emb, bnorm, counts, sse);

    const size_t lds_bytes = (size_t)(VQ_K * LDS_STRIDE + VQ_K) * sizeof(float); // ~283 KB
    vq_main<<<VQ_N / 128, 256, lds_bytes, stream>>>(x, emb, bnorm, idx, counts);

    vq_gather<<<VQ_NUMEL / 4 / 256, 256, 0, stream>>>(x, emb, idx, out, sse);

    vq_final<<<1, 1024, 0, stream>>>(counts, sse, out + VQ_NUMEL);
}